// TrustworthyNet_cluster_31628139168172
// MI455X (gfx1250) — compile-verified
//
#include <hip/hip_runtime.h>
#include <hip/hip_bf16.h>

// ---------------- problem dims ----------------
#define NN    6144      // samples
#define NV    3         // views
#define NF    512       // feature dim
#define KC    10        // clusters
#define KP    16        // padded cluster dim for WMMA
#define NBLK  2         // unrolled iterations
#define NSEG  4         // split-K segments for the big GEMM
#define SEGW  (NN / NSEG)   // 1536 columns per segment
#define NTILE (NN / 16)     // 384 row tiles
#define BN_EPS 1e-5f

typedef __attribute__((ext_vector_type(2))) float v2f;
typedef __attribute__((ext_vector_type(8))) float v8f;

__device__ __forceinline__ float softplus_f(float x) {
    // stable log1p(exp(x)) == max(x,0) + log1p(exp(-|x|))
    return fmaxf(x, 0.0f) + log1pf(expf(-fabsf(x)));
}

__device__ __forceinline__ float selu_f(float x) {
    const float sc = 1.0507009873554805f;
    const float al = 1.6732632423543772f;
    return (x > 0.0f) ? sc * x : sc * al * expm1f(x);
}

// ---------------- zT init: zT[k][n] = z_init[n][k] (k<10), 0 otherwise ----------------
__global__ void tn_init_zt(const float* __restrict__ z_init, float* __restrict__ zT) {
    int n = blockIdx.x * blockDim.x + threadIdx.x;
    if (n >= NN) return;
    #pragma unroll
    for (int k = 0; k < KP; ++k)
        zT[(size_t)k * NN + n] = (k < KC) ? z_init[(size_t)n * KC + k] : 0.0f;
}

// ---------------- proj[v][n][k] = sum_f features[v][n][f] * U[v][f][k] ----------------
// 24 blocks per view (6144/256), U[v] staged in LDS (20KB)
__global__ void tn_proj(const float* __restrict__ features, const float* __restrict__ U,
                        float* __restrict__ proj) {
    __shared__ float Us[NF * KC];
    const int tid = threadIdx.x;
    const int v = blockIdx.x / 24;
    const int n = (blockIdx.x % 24) * 256 + tid;
    for (int i = tid; i < NF * KC; i += 256)
        Us[i] = U[(size_t)v * NF * KC + i];
    __syncthreads();

    const float* frow = features + ((size_t)v * NN + n) * NF;
    float acc[KC];
    #pragma unroll
    for (int k = 0; k < KC; ++k) acc[k] = 0.0f;
    for (int f = 0; f < NF; f += 4) {
        float4 x = *(const float4*)(frow + f);
        const float xs[4] = {x.x, x.y, x.z, x.w};
        #pragma unroll
        for (int j = 0; j < 4; ++j) {
            const float* urow = &Us[(f + j) * KC];
            #pragma unroll
            for (int k = 0; k < KC; ++k) acc[k] = fmaf(xs[j], urow[k], acc[k]);
        }
    }
    float* prow = proj + ((size_t)v * NN + n) * KC;
    #pragma unroll
    for (int k = 0; k < KC; ++k) prow[k] = acc[k];
}

// ---------------- Lzp[seg][v] = lap[v][:, seg] @ z[seg] via V_WMMA_F32_16X16X4_F32 ----------------
// split-K: one wave per (segment, view, 16-row tile) => NSEG*NV*NTILE = 4608 waves.
// Wave order groups a block's 8 waves on the same (seg, view): identical B addresses -> L0 hits.
__global__ void tn_gemm_lz(const float* __restrict__ lap, const float* __restrict__ zT,
                           float* __restrict__ Lzp) {
    const int wave = (blockIdx.x * blockDim.x + threadIdx.x) >> 5;
    const int lane = threadIdx.x & 31;
    const int tile = wave % NTILE;
    const int sv   = wave / NTILE;        // seg*NV + view
    const int view = sv % NV;
    const int seg  = sv / NV;
    const int lrow = lane & 15;
    const int half = lane >> 4;

    const int c0 = seg * SEGW;
    const float* pa = lap + (size_t)view * NN * NN
                          + ((size_t)(tile * 16 + lrow)) * NN + c0 + half * 2;
    const float* pb = zT + (size_t)lrow * NN + c0 + half * 2;

    v8f acc = {};
    #pragma unroll 8
    for (int c = 0; c < SEGW; c += 4) {
        v2f a = __builtin_nontemporal_load((const v2f*)(pa + c));  // lap streamed once: NT
        v2f b = *(const v2f*)(pb + c);                             // zT: L0/L2-resident
        acc = __builtin_amdgcn_wmma_f32_16x16x4_f32(
            /*neg_a=*/false, a, /*neg_b=*/false, b,
            /*c_mod=*/(short)0, acc, /*reuse_a=*/false, /*reuse_b=*/false);
    }

    if (lrow < KC) {
        float* dst = Lzp + (((size_t)sv * NN) * KC);
        #pragma unroll
        for (int r = 0; r < 8; ++r) {
            int row = tile * 16 + half * 8 + r;
            dst[(size_t)row * KC + lrow] = acc[r];
        }
    }
}

// ---------------- per-row: H, softplus, Dempster-Shafer combine, partial BN stats ----------------
__device__ __forceinline__ void tn_ds_combine(float* cur, const float* a2) {
    float S1 = 0.f, S2 = 0.f;
    #pragma unroll
    for (int k = 0; k < KC; ++k) { S1 += cur[k]; S2 += a2[k]; }
    float u1 = (float)KC / S1, u2 = (float)KC / S2;
    float b1[KC], b2[KC];
    float sb1 = 0.f, sb2 = 0.f, dot = 0.f;
    #pragma unroll
    for (int k = 0; k < KC; ++k) {
        b1[k] = (cur[k] - 1.0f) / S1;
        b2[k] = (a2[k] - 1.0f) / S2;
        sb1 += b1[k]; sb2 += b2[k]; dot += b1[k] * b2[k];
    }
    float C = sb1 * sb2 - dot;
    float inv = 1.0f / (1.0f - C);
    float u = u1 * u2 * inv;
    float S = (float)KC / u;
    #pragma unroll
    for (int k = 0; k < KC; ++k) {
        float bk = (b1[k] * b2[k] + b1[k] * u2 + b2[k] * u1) * inv;
        cur[k] = bk * S + 1.0f;
    }
}

__global__ void tn_combine(const float* __restrict__ zT, const float* __restrict__ Lzp,
                           const float* __restrict__ proj, float* __restrict__ h_out,
                           float* __restrict__ partial) {
    const int tid = threadIdx.x;
    const int n = blockIdx.x * 256 + tid;

    float cur[KC], a2[KC];
    #pragma unroll
    for (int v = 0; v < NV; ++v) {
        float* dst = (v == 0) ? cur : a2;
        #pragma unroll
        for (int k = 0; k < KC; ++k) {
            // sum split-K partials in fixed order (deterministic)
            float lz = 0.0f;
            #pragma unroll
            for (int s = 0; s < NSEG; ++s)
                lz += Lzp[(((size_t)(s * NV + v) * NN) + n) * KC + k];
            float H = zT[(size_t)k * NN + n] - lz + proj[((size_t)v * NN + n) * KC + k];
            dst[k] = softplus_f(H) + 1.0f;
        }
        if (v > 0) tn_ds_combine(cur, a2);
    }

    #pragma unroll
    for (int k = 0; k < KC; ++k) h_out[(size_t)n * KC + k] = cur[k];

    // deterministic block reduction of sum(h), sum(h^2) per k
    __shared__ float red[256 * 2 * KC];
    #pragma unroll
    for (int k = 0; k < KC; ++k) {
        red[tid * 2 * KC + k]      = cur[k];
        red[tid * 2 * KC + KC + k] = cur[k] * cur[k];
    }
    __syncthreads();
    for (int s = 128; s > 0; s >>= 1) {
        if (tid < s) {
            #pragma unroll
            for (int j = 0; j < 2 * KC; ++j)
                red[tid * 2 * KC + j] += red[(tid + s) * 2 * KC + j];
        }
        __syncthreads();
    }
    if (tid < 2 * KC) partial[blockIdx.x * 2 * KC + tid] = red[tid];
}

// ---------------- finalize batch stats ----------------
__global__ void tn_stats(const float* __restrict__ partial, float* __restrict__ stats) {
    __shared__ float tot[2 * KC];
    const int tid = threadIdx.x;
    if (tid < 2 * KC) {
        float s = 0.f;
        for (int b = 0; b < NN / 256; ++b) s += partial[b * 2 * KC + tid];
        tot[tid] = s;
    }
    __syncthreads();
    if (tid < KC) {
        float mu  = tot[tid] / (float)NN;
        float var = tot[KC + tid] / (float)NN - mu * mu;
        stats[tid]      = mu;
        stats[KC + tid] = rsqrtf(var + BN_EPS);
    }
}

// ---------------- BN + soft-threshold SELU; emit out[t] and next zT ----------------
__global__ void tn_update(const float* __restrict__ h, const float* __restrict__ stats,
                          const float* __restrict__ gamma, const float* __restrict__ beta,
                          const float* __restrict__ theta, float* __restrict__ out_t,
                          float* __restrict__ zT) {
    const int n = blockIdx.x * blockDim.x + threadIdx.x;
    if (n >= NN) return;
    const float th = theta[0];
    #pragma unroll
    for (int k = 0; k < KC; ++k) {
        float hn = (h[(size_t)n * KC + k] - stats[k]) * stats[KC + k] * gamma[k] + beta[k];
        float zv = selu_f(hn - th) - selu_f(-hn - th);
        out_t[(size_t)n * KC + k] = zv;
        zT[(size_t)k * NN + n] = zv;   // rows KC..KP-1 stay zero from tn_init_zt
    }
}

// ---------------- host launch ----------------
extern "C" void kernel_launch(void* const* d_in, const int* in_sizes, int n_in,
                              void* d_out, int out_size, void* d_ws, size_t ws_size,
                              hipStream_t stream) {
    const float* features = (const float*)d_in[0];   // [V,N,F]
    const float* lap      = (const float*)d_in[1];   // [V,N,N]
    const float* z_init   = (const float*)d_in[2];   // [N,K]
    const float* U        = (const float*)d_in[3];   // [V,F,K]
    const float* theta    = (const float*)d_in[4];   // [1]
    const float* gamma    = (const float*)d_in[5];   // [K]
    const float* beta     = (const float*)d_in[6];   // [K]
    float* out = (float*)d_out;                      // [BLOCK,N,K]

    float* w = (float*)d_ws;
    float* zT      = w;                                   // KP*NN
    float* Lzp     = zT + (size_t)KP * NN;                // NSEG*NV*NN*KC
    float* proj    = Lzp + (size_t)NSEG * NV * NN * KC;   // NV*NN*KC
    float* hbuf    = proj + (size_t)NV * NN * KC;         // NN*KC
    float* partial = hbuf + (size_t)NN * KC;              // (NN/256)*2*KC
    float* stats   = partial + (size_t)(NN / 256) * 2 * KC; // 2*KC

    const int nblk256 = NN / 256;                      // 24
    tn_init_zt<<<nblk256, 256, 0, stream>>>(z_init, zT);
    tn_proj<<<NV * nblk256, 256, 0, stream>>>(features, U, proj);

    const int gemm_blocks = (NSEG * NV * NTILE * 32) / 256;   // 4608 waves / 8 = 576
    for (int t = 0; t < NBLK; ++t) {
        tn_gemm_lz<<<gemm_blocks, 256, 0, stream>>>(lap, zT, Lzp);
        tn_combine<<<nblk256, 256, 0, stream>>>(zT, Lzp, proj, hbuf, partial);
        tn_stats<<<1, 64, 0, stream>>>(partial, stats);
        tn_update<<<nblk256, 256, 0, stream>>>(hbuf, stats, gamma, beta, theta,
                                               out + (size_t)t * NN * KC, zT);
    }
}